// BDHLayer_63419487092893
// MI455X (gfx1250) — compile-verified
//
#include <hip/hip_runtime.h>

// Problem sizes (fixed by the reference).
#define BB 32
#define TT 1024
#define NN 128
#define CHUNK 16
#define NCHUNK (TT / CHUNK)   // 64 sequential chunks
#define NTHREAD 256           // 8 wave32 waves
// Padded LDS strides (floats): keep 16B alignment for float4 copies and
// spread per-lane row-strided fragment loads across banks.
#define WS 132                // W row stride
#define XS 132                // X-chunk row stride
#define CS 20                 // coefficient-matrix row stride

typedef __attribute__((ext_vector_type(2))) float v2f;
typedef __attribute__((ext_vector_type(8))) float v8f;

// D(16x16,f32) = A(16x4,f32) * B(4x16,f32) + C  — CDNA5 VOP3P WMMA, wave32.
__device__ __forceinline__ v8f wmma4(v2f a, v2f b, v8f c) {
  return __builtin_amdgcn_wmma_f32_16x16x4_f32(
      /*neg_a=*/false, a, /*neg_b=*/false, b,
      /*c_mod=*/(short)0, c, /*reuse_a=*/false, /*reuse_b=*/false);
}

// One workgroup per batch. W lives in LDS for the whole scan.
// Recurrence W_s = g_s*W_{s-1} + e_s*x_s x_s^T is evaluated chunk-wise:
//   y_t      = G_t * (W_prev x_t) + sum_{s<t} e_s*(G_t/G_{s+1}) x_s (x_s . x_t)
//   W_next   = G_C * W_prev + X^T diag(f) X,  f_s = e_s*G_C/G_{s+1}
// with G_t = prod_{u<t} g_u (local prefix products). All contractions use
// v_wmma_f32_16x16x4_f32.
__global__ void __launch_bounds__(NTHREAD)
hebb_scan_kernel(const float* __restrict__ x,
                 const float* __restrict__ w_init,
                 const float* __restrict__ alphap,
                 const float* __restrict__ etap,
                 const int* __restrict__ mask,
                 float* __restrict__ w_out,
                 float* __restrict__ y_out) {
  extern __shared__ float smem[];
  float* Wsh = smem;                    // NN * WS
  float* Xsh = Wsh + NN * WS;           // CHUNK * XS
  float* Csh = Xsh + CHUNK * XS;        // CHUNK * CS  (masked/decayed Gram)
  float* sG  = Csh + CHUNK * CS;        // 17 prefix products G_0..G_16
  float* sGi = sG + 17;                 // 17 reciprocals 1/G_t
  float* sE  = sGi + 17;                // 16 per-step eta*mask
  float* sF  = sE + 16;                 // 16 W-update diag weights

  const int b    = blockIdx.x;
  const int tid  = threadIdx.x;
  const int wave = tid >> 5;
  const int lane = tid & 31;
  const int half = lane >> 4;           // K / row-half selector (wave32 layout)
  const int lr   = lane & 15;
  const float alpha = *alphap;
  const float eta   = *etap;

  // ---- Load initial W into LDS (coalesced float4) ----
  const float* wsrc = w_init + (size_t)b * NN * NN;
  for (int idx = tid * 4; idx < NN * NN; idx += NTHREAD * 4) {
    const int r = idx >> 7, c = idx & 127;
    *(float4*)(&Wsh[r * WS + c]) = *(const float4*)(wsrc + idx);
  }
  __syncthreads();

  const float* xb = x + (size_t)b * TT * NN;
  const int*   mb = mask + b * TT;
  float* yb_base  = y_out + (size_t)b * TT * NN;

  const int i0 = wave * 16;             // this wave's 16-wide output column tile

  for (int ch = 0; ch < NCHUNK; ++ch) {
    const int t0 = ch * CHUNK;

    // ---- Stage X chunk (16 x 128) into LDS ----
    for (int idx4 = tid; idx4 < CHUNK * NN / 4; idx4 += NTHREAD) {
      const int r = idx4 >> 5, c4 = (idx4 & 31) << 2;
      *(float4*)(&Xsh[r * XS + c4]) = *(const float4*)(xb + (t0 + r) * NN + c4);
    }
    // Prefetch next chunk's x into cache (global_prefetch_b8).
    if (ch + 1 < NCHUNK)
      __builtin_prefetch(xb + (t0 + CHUNK) * NN + lane * 16, 0, 0);

    // ---- Per-step scalars: prefix decay products, reciprocals, gains ----
    if (tid == 0) {
      float G = 1.f;
      sG[0] = 1.f; sGi[0] = 1.f;
      for (int s = 0; s < CHUNK; ++s) {
        const int m = mb[t0 + s];
        sE[s] = m ? eta : 0.f;
        G *= m ? alpha : 1.f;
        sG[s + 1] = G;
        sGi[s + 1] = 1.f / G;           // g in {alpha,1}: never 0
      }
      for (int s = 0; s < CHUNK; ++s)
        sF[s] = sE[s] * sG[CHUNK] * sGi[s + 1];
    }
    __syncthreads();

    // ---- Inter-chunk GEMM: acc = X(16x128) * W^T tile (K=128) ----
    // Wave 0 fuses the Gram matrix S = X X^T into the same loop, reusing
    // the already-loaded A fragments (wave-uniform branch: EXEC all-ones).
    v8f acc = {};
    v8f sgram = {};
    #pragma unroll 8
    for (int k0 = 0; k0 < NN; k0 += 4) {
      v2f a  = *(const v2f*)(&Xsh[lr * XS + k0 + 2 * half]);          // A[t][k]
      v2f bw = *(const v2f*)(&Wsh[(i0 + lr) * WS + k0 + 2 * half]);   // B[k][i]=W[i][k]
      acc = wmma4(a, bw, acc);
      if (wave == 0) sgram = wmma4(a, a, sgram);                      // S[t][s]
    }
    // Row-wise decay: y_inter scaled by G_t before intra terms are added.
    #pragma unroll
    for (int v = 0; v < 8; ++v) acc[v] *= sG[v + half * 8];

    // ---- Causal/decay coefficient matrix (wave 0 stores C = coef .* S) ----
    if (wave == 0) {
      #pragma unroll
      for (int v = 0; v < 8; ++v) {
        const int m = v + half * 8;       // t (row)
        const int n = lr;                 // s (col)
        const float coef = (n < m) ? sE[n] * sG[m] * sGi[n + 1] : 0.f;
        Csh[m * CS + n] = coef * sgram[v];
      }
    }
    __syncthreads();

    // ---- This wave's X column fragments (B-layout: rows=s, col=i0+lr) ----
    // Shared by the intra-chunk GEMM (as B) and the W update (as scaled A).
    v2f xcol[4];
    #pragma unroll
    for (int q = 0; q < 4; ++q) {
      const int ks = q * 4 + 2 * half;
      xcol[q].x = Xsh[ks * XS + i0 + lr];
      xcol[q].y = Xsh[(ks + 1) * XS + i0 + lr];
    }

    // ---- Intra-chunk GEMM: acc += C(16x16) * X tile (K=16) ----
    #pragma unroll
    for (int q = 0; q < 4; ++q) {
      v2f a = *(const v2f*)(&Csh[lr * CS + q * 4 + 2 * half]);
      acc = wmma4(a, xcol[q], acc);
    }
    // ---- Write y chunk ----
    float* yb = yb_base + (size_t)t0 * NN;
    #pragma unroll
    for (int v = 0; v < 8; ++v)
      yb[(v + half * 8) * NN + i0 + lr] = acc[v];

    // ---- W update: W = G_C*W + X^T diag(f) X  (this wave's 16-row band) ----
    // A fragments depend only on i0: hoist out of the jt loop.
    v2f afr[4];
    #pragma unroll
    for (int q = 0; q < 4; ++q) {
      const int ks = q * 4 + 2 * half;
      afr[q].x = sF[ks]     * xcol[q].x;  // A[i][s] = f_s * X[s][i]
      afr[q].y = sF[ks + 1] * xcol[q].y;
    }
    const float Gend = sG[CHUNK];
    for (int jt = 0; jt < 8; ++jt) {
      const int j0 = jt * 16;
      v8f wacc;
      #pragma unroll
      for (int v = 0; v < 8; ++v)
        wacc[v] = Gend * Wsh[(i0 + v + half * 8) * WS + j0 + lr];
      #pragma unroll
      for (int q = 0; q < 4; ++q) {
        const int ks = q * 4 + 2 * half;
        v2f bx;
        bx.x = Xsh[ks * XS       + j0 + lr];   // B[s][j] = X[s][j]
        bx.y = Xsh[(ks + 1) * XS + j0 + lr];
        wacc = wmma4(afr[q], bx, wacc);
      }
      #pragma unroll
      for (int v = 0; v < 8; ++v)
        Wsh[(i0 + v + half * 8) * WS + j0 + lr] = wacc[v];
    }
    __syncthreads();   // Xsh/Csh reuse + W visible for next chunk
  }

  // ---- Store final W ----
  float* wdst = w_out + (size_t)b * NN * NN;
  for (int idx = tid * 4; idx < NN * NN; idx += NTHREAD * 4) {
    const int r = idx >> 7, c = idx & 127;
    *(float4*)(wdst + idx) = *(const float4*)(&Wsh[r * WS + c]);
  }
}

extern "C" void kernel_launch(void* const* d_in, const int* in_sizes, int n_in,
                              void* d_out, int out_size, void* d_ws, size_t ws_size,
                              hipStream_t stream) {
  const float* x      = (const float*)d_in[0];
  const float* w_init = (const float*)d_in[1];
  const float* alpha  = (const float*)d_in[2];
  const float* eta    = (const float*)d_in[3];
  const int*   maskp  = (const int*)d_in[4];

  float* w_out = (float*)d_out;                        // [B,N,N] first
  float* y_out = (float*)d_out + (size_t)BB * NN * NN; // then [B,T,N]

  const size_t smem_bytes =
      (size_t)(NN * WS + CHUNK * XS + CHUNK * CS + 17 + 17 + 16 + 16) *
      sizeof(float);
  // ~77.5 KB dynamic LDS (>64KB default) — raise the limit; 320KB/WGP on CDNA5.
  (void)hipFuncSetAttribute((const void*)hebb_scan_kernel,
                            hipFuncAttributeMaxDynamicSharedMemorySize,
                            (int)smem_bytes);

  hebb_scan_kernel<<<BB, NTHREAD, smem_bytes, stream>>>(
      x, w_init, alpha, eta, maskp, w_out, y_out);
}